// HGDC_9294309229062
// MI455X (gfx1250) — compile-verified
//
#include <hip/hip_runtime.h>

// H = hidden = 64 everywhere; Fout of every GEMM is 64.
#define HID 64

typedef __attribute__((ext_vector_type(2))) float v2f;
typedef __attribute__((ext_vector_type(8))) float v8f;

// ---------------- utility kernels ----------------

__global__ void k_zero(float* __restrict__ p, long long n) {
  long long i = (long long)blockIdx.x * blockDim.x + threadIdx.x;
  if (i < n) p[i] = 0.0f;
}

// deg[dst] += 1 for each edge (dst = second row of edge_index)
__global__ void k_degree(const int* __restrict__ ei, float* __restrict__ deg, int E) {
  int e = blockIdx.x * blockDim.x + threadIdx.x;
  if (e < E) unsafeAtomicAdd(&deg[ei[E + e]], 1.0f);
}

// deg -> deg^{-1/2} (0 if deg == 0), in place
__global__ void k_dinv(float* __restrict__ d, long long n) {
  long long i = (long long)blockIdx.x * blockDim.x + threadIdx.x;
  if (i < n) {
    float v = d[i];
    d[i] = (v > 0.0f) ? rsqrtf(v) : 0.0f;
  }
}

// R[i, 0:64] = b1[c], R[i, 64:128] = b2[c-64]  (init concat accumulator with bias)
__global__ void k_init_bias128(float* __restrict__ R, const float* __restrict__ b1,
                               const float* __restrict__ b2, long long n128) {
  long long i = (long long)blockIdx.x * blockDim.x + threadIdx.x;
  if (i < n128) {
    int c = (int)(i & 127);
    R[i] = (c < HID) ? b1[c] : b2[c - HID];
  }
}

// ---------------- fp32 WMMA GEMM: C[n x 64] = A[n x fin] * W[fin x 64] (+bias)(+relu)
// One wave computes one 16-row tile across all 4 column tiles of the 64-wide output.
// Uses V_WMMA_F32_16X16X4_F32 (exact fp32 path; GEMM cost is negligible vs scatter).
__global__ void k_gemm_wmma(const float* __restrict__ A, const float* __restrict__ W,
                            const float* __restrict__ bias, float* __restrict__ C,
                            int n_rows, int fin, int do_relu) {
  int gwave = (int)((blockIdx.x * (long long)blockDim.x + threadIdx.x) >> 5);
  int lane  = threadIdx.x & 31;
  int row0  = gwave << 4;                 // 16 rows per wave; n_rows % 16 == 0
  if (row0 >= n_rows) return;             // wave-uniform exit: EXEC all-ones in WMMA

  int m  = lane & 15;                     // row-within-tile (A) / column (B, C)
  int kh = (lane >> 4) << 1;              // lanes 0-15 -> K+0/K+1 ; lanes 16-31 -> K+2/K+3

  v8f acc[4] = {};                        // 4 col tiles x 8 accum VGPRs

  for (int k = 0; k < fin; k += 4) {
    // A fragment: 16x4 fp32, §7.12.2 layout
    const float* ap = A + (size_t)(row0 + m) * fin + (k + kh);
    v2f a;
    a.x = ap[0];
    a.y = ap[1];
    const float* wrow = W + (size_t)(k + kh) * HID;   // rows k+kh, k+kh+1 of W
#pragma unroll
    for (int t = 0; t < 4; ++t) {
      // B fragment: 4x16 fp32, mirrored layout (row striped across lanes)
      v2f b;
      b.x = wrow[t * 16 + m];
      b.y = wrow[HID + t * 16 + m];
      acc[t] = __builtin_amdgcn_wmma_f32_16x16x4_f32(
          false, a, false, b, (short)0, acc[t], false, false);
    }
  }

  // C/D layout: VGPR i holds (row0 + i + 8*(lane>=16), col = t*16 + (lane&15))
  int rbase = row0 + ((lane >> 4) << 3);
#pragma unroll
  for (int t = 0; t < 4; ++t) {
    int col  = t * 16 + m;
    float bv = bias ? bias[col] : 0.0f;
#pragma unroll
    for (int i = 0; i < 8; ++i) {
      float v = acc[t][i] + bv;
      if (do_relu) v = fmaxf(v, 0.0f);
      C[(size_t)(rbase + i) * HID + col] = v;
    }
  }
}

// ---------------- edge aggregation: out[dst, ocol:ocol+64] += norm * xw[src, :]
// 16 threads per edge, float4 per thread; hardware fp32 atomics land in L2
// (accumulator buffers are 51 MB << 192 MB global L2).
__global__ void k_scatter(const float* __restrict__ xw, const int* __restrict__ ei,
                          const float* __restrict__ dinv, float* __restrict__ out,
                          int E, int ostride, int ocol) {
  long long t = (long long)blockIdx.x * blockDim.x + threadIdx.x;
  if (t >= (long long)E * 16) return;
  int e = (int)(t >> 4);
  int c = ((int)t & 15) << 2;
  int src = ei[e];
  int dst = ei[E + e];
  float nrm = dinv[src] * dinv[dst];
  float4 v = *(const float4*)(xw + (size_t)src * HID + c);
  float* o = out + (size_t)dst * ostride + ocol + c;
  unsafeAtomicAdd(o + 0, v.x * nrm);
  unsafeAtomicAdd(o + 1, v.y * nrm);
  unsafeAtomicAdd(o + 2, v.z * nrm);
  unsafeAtomicAdd(o + 3, v.w * nrm);
}

// ---------------- readout: out[i] (+)= w * (dot(R[i,:], Wr) + br)
__global__ void k_readout(const float* __restrict__ R, const float* __restrict__ Wr,
                          const float* __restrict__ br, const float* __restrict__ w,
                          float* __restrict__ out, int n, int dim, int accum) {
  int i = blockIdx.x * blockDim.x + threadIdx.x;
  if (i >= n) return;
  const float* r = R + (size_t)i * dim;
  float acc = 0.0f;
#pragma unroll 8
  for (int c = 0; c < dim; ++c) acc += r[c] * Wr[c];
  float v = (acc + br[0]) * w[0];
  out[i] = accum ? (out[i] + v) : v;
}

// ---------------- host side ----------------

extern "C" void kernel_launch(void* const* d_in, const int* in_sizes, int n_in,
                              void* d_out, int out_size, void* d_ws, size_t ws_size,
                              hipStream_t stream) {
  (void)n_in; (void)out_size; (void)ws_size;
  const float* x   = (const float*)d_in[0];
  const int*   ei1 = (const int*)d_in[1];
  const int*   ei2 = (const int*)d_in[2];
  const float* W1   = (const float*)d_in[3];  const float* b1   = (const float*)d_in[4];
  const float* Wc11 = (const float*)d_in[5];  const float* bc11 = (const float*)d_in[6];
  const float* Wc12 = (const float*)d_in[7];  const float* bc12 = (const float*)d_in[8];
  const float* Wc21 = (const float*)d_in[9];  const float* bc21 = (const float*)d_in[10];
  const float* Wc22 = (const float*)d_in[11]; const float* bc22 = (const float*)d_in[12];
  const float* Wc31 = (const float*)d_in[13]; const float* bc31 = (const float*)d_in[14];
  const float* Wc32 = (const float*)d_in[15]; const float* bc32 = (const float*)d_in[16];
  const float* Wr0 = (const float*)d_in[17];  const float* br0 = (const float*)d_in[18];
  const float* Wr1 = (const float*)d_in[19];  const float* br1 = (const float*)d_in[20];
  const float* Wr2 = (const float*)d_in[21];  const float* br2 = (const float*)d_in[22];
  const float* Wr3 = (const float*)d_in[23];  const float* br3 = (const float*)d_in[24];
  const float* w0 = (const float*)d_in[25];
  const float* w1 = (const float*)d_in[26];
  const float* w2 = (const float*)d_in[27];
  const float* w3 = (const float*)d_in[28];

  const int N = in_sizes[0] / HID;       // 100000
  const int E = in_sizes[1] / 2;         // 1600000
  float* out = (float*)d_out;

  // workspace layout (floats); each chunk is 16B-aligned (N*4 bytes = 400000B)
  float* ws    = (float*)d_ws;
  float* dinv1 = ws;                          // N
  float* dinv2 = dinv1 + (size_t)N;           // N
  float* R0    = dinv2 + (size_t)N;           // N*64
  float* R1    = R0 + (size_t)N * HID;        // N*128
  float* R2    = R1 + (size_t)N * 2 * HID;    // N*128
  float* R3    = R2 + (size_t)N * 2 * HID;    // N*128
  float* tmpA  = R3 + (size_t)N * 2 * HID;    // N*64
  float* tmpB  = tmpA + (size_t)N * HID;      // N*64

  const int BLK = 256;
  auto blocks = [&](long long n) { return (unsigned)((n + BLK - 1) / BLK); };
  const long long n2     = 2LL * N;
  const long long n128   = (long long)N * 2 * HID;
  const long long escat  = (long long)E * 16;
  const unsigned gemmGrid = blocks(((long long)((N + 15) / 16)) * 32); // 1 wave / 16 rows

  // degrees -> dinv for both edge sets
  k_zero<<<blocks(n2), BLK, 0, stream>>>(dinv1, n2);
  k_degree<<<blocks(E), BLK, 0, stream>>>(ei1, dinv1, E);
  k_degree<<<blocks(E), BLK, 0, stream>>>(ei2, dinv2, E);
  k_dinv<<<blocks(n2), BLK, 0, stream>>>(dinv1, n2);

  // R0 = relu(x @ W1 + b1)
  k_gemm_wmma<<<gemmGrid, BLK, 0, stream>>>(x, W1, b1, R0, N, HID, 1);
  k_readout<<<blocks(N), BLK, 0, stream>>>(R0, Wr0, br0, w0, out, N, HID, 0);

  // layer 1: R1 = concat(conv(R0,Wc11,e1), conv(R0,Wc12,e2))
  k_gemm_wmma<<<gemmGrid, BLK, 0, stream>>>(R0, Wc11, nullptr, tmpA, N, HID, 0);
  k_gemm_wmma<<<gemmGrid, BLK, 0, stream>>>(R0, Wc12, nullptr, tmpB, N, HID, 0);
  k_init_bias128<<<blocks(n128), BLK, 0, stream>>>(R1, bc11, bc12, n128);
  k_scatter<<<blocks(escat), BLK, 0, stream>>>(tmpA, ei1, dinv1, R1, E, 2 * HID, 0);
  k_scatter<<<blocks(escat), BLK, 0, stream>>>(tmpB, ei2, dinv2, R1, E, 2 * HID, HID);
  k_readout<<<blocks(N), BLK, 0, stream>>>(R1, Wr1, br1, w1, out, N, 2 * HID, 1);

  // layer 2
  k_gemm_wmma<<<gemmGrid, BLK, 0, stream>>>(R1, Wc21, nullptr, tmpA, N, 2 * HID, 0);
  k_gemm_wmma<<<gemmGrid, BLK, 0, stream>>>(R1, Wc22, nullptr, tmpB, N, 2 * HID, 0);
  k_init_bias128<<<blocks(n128), BLK, 0, stream>>>(R2, bc21, bc22, n128);
  k_scatter<<<blocks(escat), BLK, 0, stream>>>(tmpA, ei1, dinv1, R2, E, 2 * HID, 0);
  k_scatter<<<blocks(escat), BLK, 0, stream>>>(tmpB, ei2, dinv2, R2, E, 2 * HID, HID);
  k_readout<<<blocks(N), BLK, 0, stream>>>(R2, Wr2, br2, w2, out, N, 2 * HID, 1);

  // layer 3
  k_gemm_wmma<<<gemmGrid, BLK, 0, stream>>>(R2, Wc31, nullptr, tmpA, N, 2 * HID, 0);
  k_gemm_wmma<<<gemmGrid, BLK, 0, stream>>>(R2, Wc32, nullptr, tmpB, N, 2 * HID, 0);
  k_init_bias128<<<blocks(n128), BLK, 0, stream>>>(R3, bc31, bc32, n128);
  k_scatter<<<blocks(escat), BLK, 0, stream>>>(tmpA, ei1, dinv1, R3, E, 2 * HID, 0);
  k_scatter<<<blocks(escat), BLK, 0, stream>>>(tmpB, ei2, dinv2, R3, E, 2 * HID, HID);
  k_readout<<<blocks(N), BLK, 0, stream>>>(R3, Wr3, br3, w3, out, N, 2 * HID, 1);
}